// PredictionHeadEdge_2104533975131
// MI455X (gfx1250) — compile-verified
//
#include <hip/hip_runtime.h>
#include <hip/hip_bf16.h>

// ---------------------------------------------------------------------------
// PredictionHeadEdge for gfx1250 (MI455X, wave32, WMMA).
// All GEMMs use V_WMMA_F32_16X16X4_F32 (exact fp32 matrix op).
//
// Fragment layout (CDNA5 ISA 7.12.2, wave32):
//   A (16x4 f32, 2 VGPR):  lane L holds A[L&15][2*(L>>4) + {0,1}]
//   B (4x16 f32, 2 VGPR):  lane L holds B[2*(L>>4) + {0,1}][L&15]
//   C/D (16x16 f32, 8 VGPR): elem t of lane L = D[t + 8*(L>>4)][L&15]
// ---------------------------------------------------------------------------

typedef __attribute__((ext_vector_type(2))) float v2f;
typedef __attribute__((ext_vector_type(8))) float v8f;

#define NNODE 1024
#define NEDGE 32768
#define SD    256
#define VD    64
#define ED    128
#define NGR   16
#define NATM  16
#define NBND  5

__device__ __forceinline__ v8f wmma_f32_k4(v2f a, v2f b, v8f c) {
  // (neg_a, A, neg_b, B, c_mod, C, reuse_a, reuse_b)
  return __builtin_amdgcn_wmma_f32_16x16x4_f32(false, a, false, b, (short)0, c,
                                               false, false);
}

// silu(x) = x * sigmoid(x); v_rcp_f32 (~1 ulp) instead of the IEEE division
// ladder (v_div_scale/v_div_fmas/v_div_fixup) — halves epilogue VALU work.
__device__ __forceinline__ float silu_f(float x) {
  return x * __builtin_amdgcn_rcpf(1.0f + __expf(-x));
}

// --- scratch init: map = -1 ------------------------------------------------
__global__ void k_init(int* __restrict__ map) {
  int idx = blockIdx.x * blockDim.x + threadIdx.x;
  if (idx < NNODE * NNODE) map[idx] = -1;
}

// --- map[j*n+i] = max edge id (deterministic "last set wins") --------------
__global__ void k_map(const int* __restrict__ ei, int* __restrict__ map) {
  int g = blockIdx.x * blockDim.x + threadIdx.x;
  if (g >= NEDGE) return;
  int j = ei[g], i = ei[NEDGE + g];
  atomicMax(&map[j * NNODE + i], g);
}

// --- e_sym[g,k] = 0.5*(e[map[j,i],k] + (map[i,j]>=0 ? e[map[i,j],k] : 0)) --
__global__ void k_esym(const float* __restrict__ e, const int* __restrict__ ei,
                       const int* __restrict__ map, float* __restrict__ esym) {
  int idx = blockIdx.x * blockDim.x + threadIdx.x;
  if (idx >= NEDGE * ED) return;
  int g = idx >> 7;           // ED == 128
  int k = idx & (ED - 1);
  int j = ei[g], i = ei[NEDGE + g];
  int fwd = map[j * NNODE + i];
  int rev = map[i * NNODE + j];
  float vf = e[fwd * ED + k];
  float vr = (rev >= 0) ? e[rev * ED + k] : 0.0f;
  esym[idx] = 0.5f * (vf + vr);
}

// --- s1 = silu(s @ W_shared^T + b_shared), one wave per 16x16 tile ---------
__global__ void k_sgemm(const float* __restrict__ s, const float* __restrict__ W,
                        const float* __restrict__ bias, float* __restrict__ s1) {
  int wave = threadIdx.x >> 5;
  int lane = threadIdx.x & 31;
  int tile = blockIdx.x * (blockDim.x >> 5) + wave;   // 1024 tiles
  int row0 = (tile >> 4) * 16;
  int col0 = (tile & 15) * 16;
  int mn = lane & 15, half = lane >> 4;

  v8f acc = {};
  const float* arow = s + (size_t)(row0 + mn) * SD;
  const float* brow = W + (size_t)(col0 + mn) * SD;
#pragma unroll 8
  for (int kb = 0; kb < SD; kb += 4) {
    v2f a, b;
    a.x = arow[kb + 2 * half]; a.y = arow[kb + 2 * half + 1];
    b.x = brow[kb + 2 * half]; b.y = brow[kb + 2 * half + 1];
    acc = wmma_f32_k4(a, b, acc);
  }
  int col = col0 + mn;
  float bv = bias[col];
#pragma unroll
  for (int t = 0; t < 8; ++t) {
    int m = t + 8 * half;
    s1[(size_t)(row0 + m) * SD + col] = silu_f(acc[t] + bv);
  }
}

// --- atoms_pred = s1 @ W_atoms^T + b_atoms (single 16-col tile, NA==16) ----
__global__ void k_atoms(const float* __restrict__ s1, const float* __restrict__ W,
                        const float* __restrict__ bias, float* __restrict__ out) {
  int wave = threadIdx.x >> 5;
  int lane = threadIdx.x & 31;
  int row0 = (blockIdx.x * (blockDim.x >> 5) + wave) * 16;   // 64 tiles
  int mn = lane & 15, half = lane >> 4;

  v8f acc = {};
  const float* arow = s1 + (size_t)(row0 + mn) * SD;
  const float* brow = W + (size_t)mn * SD;
#pragma unroll 8
  for (int kb = 0; kb < SD; kb += 4) {
    v2f a, b;
    a.x = arow[kb + 2 * half]; a.y = arow[kb + 2 * half + 1];
    b.x = brow[kb + 2 * half]; b.y = brow[kb + 2 * half + 1];
    acc = wmma_f32_k4(a, b, acc);
  }
  float bv = bias[mn];
#pragma unroll
  for (int t = 0; t < 8; ++t) {
    int m = t + 8 * half;
    out[(size_t)(row0 + m) * NATM + mn] = acc[t] + bv;
  }
}

// --- coords_raw = p + v @ W_coords^T ---------------------------------------
__global__ void k_coords1(const float* __restrict__ v, const float* __restrict__ p,
                          const float* __restrict__ Wc, float* __restrict__ craw) {
  int n = blockIdx.x * blockDim.x + threadIdx.x;
  if (n >= NNODE) return;
#pragma unroll
  for (int x = 0; x < 3; ++x) {
    float acc = p[n * 3 + x];
    const float* vr = v + (size_t)n * 3 * VD + x * VD;
#pragma unroll 8
    for (int k = 0; k < VD; ++k) acc = fmaf(vr[k], Wc[k], acc);
    craw[n * 3 + x] = acc;
  }
}

// --- deterministic per-graph mean (single block, LDS tree reduction) -------
__global__ void k_segmean(const float* __restrict__ craw,
                          const int* __restrict__ batch,
                          float* __restrict__ mean) {
  __shared__ float part[128 * 64];   // 128 threads x (16 graphs * 4 bins) = 32KB
  int tid = threadIdx.x;
  for (int b = 0; b < 64; ++b) part[tid * 64 + b] = 0.0f;
  for (int n = tid; n < NNODE; n += 128) {
    int g = batch[n];
    part[tid * 64 + g * 4 + 0] += craw[n * 3 + 0];
    part[tid * 64 + g * 4 + 1] += craw[n * 3 + 1];
    part[tid * 64 + g * 4 + 2] += craw[n * 3 + 2];
    part[tid * 64 + g * 4 + 3] += 1.0f;
  }
  __syncthreads();
  for (int s2 = 64; s2 > 0; s2 >>= 1) {
    if (tid < s2)
      for (int b = 0; b < 64; ++b)
        part[tid * 64 + b] += part[(tid + s2) * 64 + b];
    __syncthreads();
  }
  if (tid < NGR) {
    float cnt = fmaxf(part[tid * 4 + 3], 1.0f);
    float rc = 1.0f / cnt;   // exact-ish; 16 values only, precision matters
    mean[tid * 3 + 0] = part[tid * 4 + 0] * rc;
    mean[tid * 3 + 1] = part[tid * 4 + 1] * rc;
    mean[tid * 3 + 2] = part[tid * 4 + 2] * rc;
  }
}

// --- coords_fin = coords_raw - mean[batch] ; write output ------------------
__global__ void k_coords2(const float* __restrict__ craw, const int* __restrict__ batch,
                          const float* __restrict__ mean, float* __restrict__ cfin,
                          float* __restrict__ out) {
  int idx = blockIdx.x * blockDim.x + threadIdx.x;
  if (idx >= NNODE * 3) return;
  int n = idx / 3, x = idx - n * 3;
  float val = craw[idx] - mean[batch[n] * 3 + x];
  cfin[idx] = val;
  out[idx] = val;
}

// --- d[g] = |c[i]-c[j]|^2 --------------------------------------------------
__global__ void k_dist(const float* __restrict__ cfin, const int* __restrict__ ei,
                       float* __restrict__ de) {
  int g = blockIdx.x * blockDim.x + threadIdx.x;
  if (g >= NEDGE) return;
  int j = ei[g], i = ei[NEDGE + g];
  float d = 0.0f;
#pragma unroll
  for (int x = 0; x < 3; ++x) {
    float t = cfin[i * 3 + x] - cfin[j * 3 + x];
    d = fmaf(t, t, d);
  }
  de[g] = d;
}

// --- fused bond MLP: 16 edges per block, 16 waves, 3 chained WMMA GEMMs ----
__global__ void k_bonds(const float* __restrict__ esym, const float* __restrict__ s1,
                        const int* __restrict__ ei, const float* __restrict__ de,
                        const float* __restrict__ Wbm, const float* __restrict__ bbm,
                        const float* __restrict__ Wb0, const float* __restrict__ bb0,
                        const float* __restrict__ Wb1, const float* __restrict__ bb1,
                        float* __restrict__ out) {
  __shared__ float ebuf[16][260];   // f (256 cols) + d at col 256
  __shared__ float hbuf[16][260];   // silu hidden
  __shared__ int sj[16], si[16];

  int tid = threadIdx.x;
  int wave = tid >> 5, lane = tid & 31;
  int mn = lane & 15, half = lane >> 4;
  int e0 = blockIdx.x * 16;

  if (tid < 16) {
    sj[tid] = ei[e0 + tid];
    si[tid] = ei[NEDGE + e0 + tid];
    ebuf[tid][256] = de[e0 + tid];
  }
  __syncthreads();

  int col0 = wave * 16;
  int col = col0 + mn;

  // GEMM1: f = e_sym @ W_bond_map^T   (K = 128)
  v8f acc1 = {};
  {
    const float* arow = esym + (size_t)(e0 + mn) * ED;
    const float* brow = Wbm + (size_t)col * ED;
#pragma unroll 4
    for (int kb = 0; kb < ED; kb += 4) {
      v2f a, b;
      a.x = arow[kb + 2 * half]; a.y = arow[kb + 2 * half + 1];
      b.x = brow[kb + 2 * half]; b.y = brow[kb + 2 * half + 1];
      acc1 = wmma_f32_k4(a, b, acc1);
    }
  }
  {
    float bv = bbm[col];
#pragma unroll
    for (int t = 0; t < 8; ++t) {
      int m = t + 8 * half;
      ebuf[m][col] = acc1[t] + bv + s1[(size_t)si[m] * SD + col]
                              + s1[(size_t)sj[m] * SD + col];
    }
  }
  __syncthreads();

  // GEMM2: h = silu([f, d] @ W_bonds0^T + b_bonds0)   (K = 257)
  v8f acc2 = {};
  {
    const float* brow = Wb0 + (size_t)col * (SD + 1);
#pragma unroll 4
    for (int kb = 0; kb < SD; kb += 4) {
      v2f a, b;
      a.x = ebuf[mn][kb + 2 * half]; a.y = ebuf[mn][kb + 2 * half + 1];
      b.x = brow[kb + 2 * half];     b.y = brow[kb + 2 * half + 1];
      acc2 = wmma_f32_k4(a, b, acc2);
    }
    float wlast = brow[SD];      // k = 256 column multiplies d
    float bv = bb0[col];
#pragma unroll
    for (int t = 0; t < 8; ++t) {
      int m = t + 8 * half;
      hbuf[m][col] = silu_f(acc2[t] + ebuf[m][256] * wlast + bv);
    }
  }
  __syncthreads();

  // GEMM3: bonds = h @ W_bonds1^T + b_bonds1 (N padded 5 -> 16), wave 0 only
  if (wave == 0) {
    v8f acc3 = {};
    const float* brow = Wb1 + (size_t)((mn < NBND) ? mn : (NBND - 1)) * SD;
#pragma unroll 4
    for (int kb = 0; kb < SD; kb += 4) {
      v2f a, b;
      a.x = hbuf[mn][kb + 2 * half]; a.y = hbuf[mn][kb + 2 * half + 1];
      b.x = (mn < NBND) ? brow[kb + 2 * half]     : 0.0f;
      b.y = (mn < NBND) ? brow[kb + 2 * half + 1] : 0.0f;
      acc3 = wmma_f32_k4(a, b, acc3);
    }
    if (mn < NBND) {
      float bv = bb1[mn];
#pragma unroll
      for (int t = 0; t < 8; ++t) {
        int m = t + 8 * half;
        out[(size_t)(e0 + m) * NBND + mn] = acc3[t] + bv;
      }
    }
  }
}

extern "C" void kernel_launch(void* const* d_in, const int* in_sizes, int n_in,
                              void* d_out, int out_size, void* d_ws, size_t ws_size,
                              hipStream_t stream) {
  (void)in_sizes; (void)n_in; (void)out_size; (void)ws_size;

  const float* s    = (const float*)d_in[0];
  const float* v    = (const float*)d_in[1];
  const float* p    = (const float*)d_in[2];
  const float* e    = (const float*)d_in[3];
  const int*   batch= (const int*)d_in[4];
  const int*   ei   = (const int*)d_in[5];
  const float* Wsh  = (const float*)d_in[6];
  const float* bsh  = (const float*)d_in[7];
  const float* Wbm  = (const float*)d_in[8];
  const float* bbm  = (const float*)d_in[9];
  const float* Wb0  = (const float*)d_in[10];
  const float* bb0  = (const float*)d_in[11];
  const float* Wb1  = (const float*)d_in[12];
  const float* bb1  = (const float*)d_in[13];
  const float* Wc   = (const float*)d_in[14];
  const float* Wat  = (const float*)d_in[15];
  const float* bat  = (const float*)d_in[16];

  // workspace layout (bytes)
  char* ws = (char*)d_ws;
  float* s1   = (float*)(ws);                              //  1 MB
  float* esym = (float*)(ws + (size_t)(1 << 20));          // 16 MB
  int*   map  = (int*)  (ws + (size_t)(17 << 20));         //  4 MB
  float* craw = (float*)(ws + (size_t)(21 << 20));         // 12 KB (pad 16K)
  float* cfin = (float*)(ws + (size_t)(21 << 20) + 16384);
  float* de   = (float*)(ws + (size_t)(21 << 20) + 32768); // 128 KB
  float* mean = (float*)(ws + (size_t)(21 << 20) + 32768 + NEDGE * 4);

  float* out_coords = (float*)d_out;            // [1024, 3]
  float* out_atoms  = out_coords + NNODE * 3;   // [1024, 16]
  float* out_bonds  = out_atoms + NNODE * NATM; // [32768, 5]

  k_init   <<<(NNODE * NNODE + 255) / 256, 256, 0, stream>>>(map);
  k_map    <<<(NEDGE + 255) / 256,          256, 0, stream>>>(ei, map);
  k_esym   <<<(NEDGE * ED + 255) / 256,     256, 0, stream>>>(e, ei, map, esym);
  k_sgemm  <<<128, 256, 0, stream>>>(s, Wsh, bsh, s1);
  k_atoms  <<<8,   256, 0, stream>>>(s1, Wat, bat, out_atoms);
  k_coords1<<<(NNODE + 255) / 256, 256, 0, stream>>>(v, p, Wc, craw);
  k_segmean<<<1, 128, 0, stream>>>(craw, batch, mean);
  k_coords2<<<(NNODE * 3 + 255) / 256, 256, 0, stream>>>(craw, batch, mean, cfin, out_coords);
  k_dist   <<<(NEDGE + 255) / 256, 256, 0, stream>>>(cfin, ei, de);
  k_bonds  <<<NEDGE / 16, 512, 0, stream>>>(esym, s1, ei, de, Wbm, bbm,
                                            Wb0, bb0, Wb1, bb1, out_bonds);
}